// BondLengthDeviation_51213190037905
// MI455X (gfx1250) — compile-verified
//
#include <hip/hip_runtime.h>

// Problem constants (from the reference setup):
//   stacked_radii: (2, 4, 512, 3) f32  -> 8 frames of 512 atoms
//   cell:          (3, 3) f32          -> 27 periodic shifts = combos @ cell
//   bonds:         (768, 2) i32
//   mean_bond_lens:(768,)  f32
// Outputs (concatenated in d_out, f32):
//   bond_lens: (2, 4, 768) = 6144 floats
//   max_bond_dev_per_replica: (4,) floats
#define NATOMS   512
#define NBONDS   768
#define NFRAMES  8
#define NREP     4

__global__ __launch_bounds__(256) void bond_len_kernel(
    const float* __restrict__ radii,   // (NFRAMES, NATOMS, 3)
    const float* __restrict__ cell,    // (3, 3)
    const int*   __restrict__ bonds,   // (NBONDS, 2)
    float*       __restrict__ out)     // (NFRAMES, NBONDS)
{
    // 6 KB of the 320 KB/WGP LDS; 16B aligned for async B128 transfers.
    __shared__ __align__(16) float s_coords[NATOMS * 3];

    const int frame = blockIdx.y;
    const float* fr = radii + frame * (NATOMS * 3);

    // CDNA5 async memory->LDS staging: 384 x B128 transfers, no VGPR
    // round-trip, tracked on ASYNCcnt. Global side: frame offset is a
    // multiple of 6144 B from a 256B-aligned allocation -> 16B aligned.
    for (int i = threadIdx.x; i < (NATOMS * 3) / 4; i += 256) {
        unsigned lds_off = (unsigned)(uintptr_t)(void*)&s_coords[4 * i];
        const float* gp = fr + 4 * i;
        asm volatile("global_load_async_to_lds_b128 %0, %1, off"
                     :: "v"(lds_off), "v"(gp)
                     : "memory");
    }
    // Wait for this wave's async transfers, then sync the workgroup so all
    // waves see the fully-staged LDS tile.
    asm volatile("s_wait_asynccnt 0x0" ::: "memory");
    __syncthreads();

    // Cell rows (uniform -> scalar loads broadcast across the wave).
    const float c00 = cell[0], c01 = cell[1], c02 = cell[2];
    const float c10 = cell[3], c11 = cell[4], c12 = cell[5];
    const float c20 = cell[6], c21 = cell[7], c22 = cell[8];

    const int b = blockIdx.x * 256 + threadIdx.x;
    if (b >= NBONDS) return;

    const int a0 = bonds[2 * b + 0];
    const int a1 = bonds[2 * b + 1];

    const float dx = s_coords[3 * a0 + 0] - s_coords[3 * a1 + 0];
    const float dy = s_coords[3 * a0 + 1] - s_coords[3 * a1 + 1];
    const float dz = s_coords[3 * a0 + 2] - s_coords[3 * a1 + 2];

    // Minimum over the 27 shifts of the squared distance; sqrt is monotone
    // and correctly rounded, so sqrt(min d2) == min(sqrt d2) bit-exactly.
    float best = 3.402823466e+38f;
#pragma unroll
    for (int sz = -1; sz <= 1; ++sz) {
#pragma unroll
        for (int sy = -1; sy <= 1; ++sy) {
#pragma unroll
            for (int sx = -1; sx <= 1; ++sx) {
                const float fsx = (float)sx, fsy = (float)sy, fsz = (float)sz;
                const float shx = fsx * c00 + fsy * c10 + fsz * c20;
                const float shy = fsx * c01 + fsy * c11 + fsz * c21;
                const float shz = fsx * c02 + fsy * c12 + fsz * c22;
                const float ex = dx - shx;
                const float ey = dy - shy;
                const float ez = dz - shz;
                const float d2 = ex * ex + ey * ey + ez * ez;
                best = fminf(best, d2);
            }
        }
    }

    out[frame * NBONDS + b] = sqrtf(best);
}

// Deterministic per-replica max reduction: 4 blocks, one per replica r.
// max over i in {0,1}, b in [0,768) of |bond_lens[(i*4+r)*768 + b] - mean[b]|
__global__ __launch_bounds__(256) void max_dev_kernel(
    const float* __restrict__ bond_lens, // (NFRAMES, NBONDS) = d_out
    const float* __restrict__ mean,      // (NBONDS,)
    float*       __restrict__ out_max)   // (NREP,)
{
    __shared__ float red[256];
    const int r = blockIdx.x;

    float m = 0.0f;  // deviations are non-negative
    for (int t = threadIdx.x; t < 2 * NBONDS; t += 256) {
        const int i = t / NBONDS;
        const int b = t - i * NBONDS;
        const float v = fabsf(bond_lens[(i * NREP + r) * NBONDS + b] - mean[b]);
        m = fmaxf(m, v);
    }
    red[threadIdx.x] = m;
    __syncthreads();

#pragma unroll
    for (int s = 128; s > 0; s >>= 1) {
        if (threadIdx.x < s)
            red[threadIdx.x] = fmaxf(red[threadIdx.x], red[threadIdx.x + s]);
        __syncthreads();
    }

    if (threadIdx.x == 0) out_max[r] = red[0];
}

extern "C" void kernel_launch(void* const* d_in, const int* in_sizes, int n_in,
                              void* d_out, int out_size, void* d_ws, size_t ws_size,
                              hipStream_t stream) {
    const float* radii = (const float*)d_in[0];   // (2,4,512,3)
    const float* cell  = (const float*)d_in[1];   // (3,3)
    const int*   bonds = (const int*)d_in[2];     // (768,2)
    const float* mean  = (const float*)d_in[3];   // (768,)
    float* out = (float*)d_out;                    // 6144 + 4 floats

    dim3 grid1((NBONDS + 255) / 256, NFRAMES);    // (3, 8) blocks
    bond_len_kernel<<<grid1, 256, 0, stream>>>(radii, cell, bonds, out);

    max_dev_kernel<<<NREP, 256, 0, stream>>>(out, mean, out + NFRAMES * NBONDS);
}